// BackBone_13804024889423
// MI455X (gfx1250) — compile-verified
//
#include <hip/hip_runtime.h>

typedef __attribute__((ext_vector_type(2))) float v2f;
typedef __attribute__((ext_vector_type(8))) float v8f;

#define T_STEPS 20
#define IN_DIM_C 340
#define HID_C 128
#define EDGE_C 32
#define TPC_C 10
#define B_TOTAL 8192
#define MROWS (B_TOTAL * T_STEPS)   // 163840 rows of the input GEMM

#define WMMA4(a, b, c) \
  __builtin_amdgcn_wmma_f32_16x16x4_f32(false, (a), false, (b), (short)0, (c), false, false)

// Ping-pong software-pipelined K loop: 8 output tiles, steps of K=4.
// Invariant: buf0 holds step k0 at loop entry; phase A loads k0+4 into buf1 and
// computes k0; phase B loads k0+8 into buf0 and computes k0+4. No register moves.
// All prefetch addresses stay in [KSTART, KTOT) by construction of the bound.
template <int KTOT, int KSTART>
__device__ __forceinline__ void wmma_kloop(const float* __restrict__ arow,
                                           const float* const* bp, v8f* acc) {
  v2f a0 = *(const v2f*)(arow + KSTART);
  v2f a1;
  v2f b0[8], b1[8];
#pragma unroll
  for (int j = 0; j < 8; ++j) b0[j] = *(const v2f*)(bp[j] + KSTART);

  int k0 = KSTART;
  for (; k0 + 8 < KTOT; k0 += 8) {
    a1 = *(const v2f*)(arow + k0 + 4);
#pragma unroll
    for (int j = 0; j < 8; ++j) b1[j] = *(const v2f*)(bp[j] + k0 + 4);
#pragma unroll
    for (int j = 0; j < 8; ++j) acc[j] = WMMA4(a0, b0[j], acc[j]);
    a0 = *(const v2f*)(arow + k0 + 8);
#pragma unroll
    for (int j = 0; j < 8; ++j) b0[j] = *(const v2f*)(bp[j] + k0 + 8);
#pragma unroll
    for (int j = 0; j < 8; ++j) acc[j] = WMMA4(a1, b1[j], acc[j]);
  }
  constexpr int NSTEPS = (KTOT - KSTART) / 4;
  if constexpr ((NSTEPS & 1) == 0) {
    // two steps remain: k0 (in buf0) and k0+4
    a1 = *(const v2f*)(arow + k0 + 4);
#pragma unroll
    for (int j = 0; j < 8; ++j) b1[j] = *(const v2f*)(bp[j] + k0 + 4);
#pragma unroll
    for (int j = 0; j < 8; ++j) acc[j] = WMMA4(a0, b0[j], acc[j]);
#pragma unroll
    for (int j = 0; j < 8; ++j) acc[j] = WMMA4(a1, b1[j], acc[j]);
  } else {
    // one step remains (in buf0)
#pragma unroll
    for (int j = 0; j < 8; ++j) acc[j] = WMMA4(a0, b0[j], acc[j]);
  }
}

// =============== Kernel 1: gx = history @ W_ih^T + (b_ih + b_hh) ===============
// 5120 blocks x 256 thr (8 waves = 2 row-groups x 4 col-strips): 32 x 512 tile.
#define G1_ROWS 32
#define XS_STRIDE 344   // 340 padded, 16B-aligned rows

__launch_bounds__(256)
__global__ void gx_gemm_kernel(const float* __restrict__ history,
                               const float* __restrict__ W_ih,
                               const float* __restrict__ b_ih,
                               const float* __restrict__ b_hh,
                               float* __restrict__ gx) {
  __shared__ __align__(16) float xs[G1_ROWS * XS_STRIDE];  // 44 KB
  const int tid  = threadIdx.x;
  const int lane = tid & 31;
  const int wave = tid >> 5;
  const int rg   = wave >> 2;            // row group 0..1
  const int cs   = wave & 3;             // col strip 0..3 (128 cols each)
  const int l16  = lane & 15;
  const int khalf = (lane >> 4) << 1;    // fragment K-pair select (0 or 2)
  const int ro    = (lane >> 4) << 3;    // D-tile row offset (+8 upper half)
  const size_t rowbase = (size_t)blockIdx.x * G1_ROWS;

  // stage A tile (32 x 340) as float4 (340 = 85*4, rows 16B aligned)
  {
    const float4* src = (const float4*)(history + rowbase * IN_DIM_C);
    float4* dst = (float4*)xs;
    for (int i = tid; i < G1_ROWS * 85; i += 256) {
      int r = i / 85, q = i - r * 85;
      dst[r * 86 + q] = src[r * 85 + q];   // XS_STRIDE/4 == 86
    }
  }
  __syncthreads();

  const float* wp[8];
  v8f acc[8];
#pragma unroll
  for (int j = 0; j < 8; ++j) {
    int n = cs * 128 + 16 * j + l16;       // gate column this lane holds
    wp[j] = W_ih + (size_t)n * IN_DIM_C + khalf;
    float bv = b_ih[n] + b_hh[n];
#pragma unroll
    for (int r = 0; r < 8; ++r) acc[j][r] = bv;
  }
  const float* xrow = xs + (rg * 16 + l16) * XS_STRIDE + khalf;

  wmma_kloop<IN_DIM_C, 0>(xrow, wp, acc);   // 85 K-steps

  // store gate pre-activations (base pointer + immediate row offsets)
#pragma unroll
  for (int j = 0; j < 8; ++j) {
    float* op = gx + (rowbase + rg * 16 + ro) * 512 + cs * 128 + 16 * j + l16;
#pragma unroll
    for (int r = 0; r < 8; ++r) op[r * 512] = acc[j][r];
  }
}

// =============== Kernel 2: LSTM recurrence, W_hh resident in LDS ===============
// 256 blocks x 256 thr (8 waves = 2 row-groups x 4 col-groups), 32 rows/block.
// Wave owns interleaved gate cols {g*128 + cw*32 + s*16 + l16} so i/f/g/o for a
// hidden unit live in ONE wave -> cell update in registers. gx pipelined over t.
#define R2_ROWS 32
#define HS_STRIDE 132
#define WHH_STRIDE 132                    // 64-bank conflict-free fragment loads
#define WHH_FLOATS (512 * WHH_STRIDE)
#define SMEM2_BYTES ((WHH_FLOATS + R2_ROWS * HS_STRIDE) * 4)   // 287232 B

__launch_bounds__(256)
__global__ void lstm_recur_kernel(const float* __restrict__ gx,
                                  const float* __restrict__ W_hh,
                                  float* __restrict__ h_last) {
  extern __shared__ float smem[];
  float* whh = smem;                       // swizzled [cw][j][c16][K]
  float* hs  = smem + WHH_FLOATS;          // [32][132]

  const int tid  = threadIdx.x;
  const int lane = tid & 31;
  const int wave = tid >> 5;
  const int rg   = wave >> 2;
  const int cw   = wave & 3;
  const int l16  = lane & 15;
  const int khalf = (lane >> 4) << 1;
  const int ro    = (lane >> 4) << 3;
  const int rowbase = blockIdx.x * R2_ROWS;

  // copy W_hh (512x128) into LDS, swizzled into per-(colgroup,tile) strips
  for (int idx = tid; idx < 512 * HID_C; idx += 256) {
    int n = idx >> 7, k = idx & 127;
    int gate = n >> 7, rem = n & 127;
    int w = rem >> 5, sub = (rem >> 4) & 1, c16 = n & 15;
    whh[((w * 8 + gate * 2 + sub) * 16 + c16) * WHH_STRIDE + k] = W_hh[idx];
  }
  for (int i = tid; i < R2_ROWS * HS_STRIDE; i += 256) hs[i] = 0.f;  // h0 = 0
  __syncthreads();

  const float* bb[8];
  const float* gp[8];
#pragma unroll
  for (int j = 0; j < 8; ++j) {
    bb[j] = whh + ((cw * 8 + j) * 16 + l16) * WHH_STRIDE + khalf;
    int gate = j >> 1, sub = j & 1;
    int gcol = gate * 128 + cw * 32 + sub * 16 + l16;
    gp[j] = gx + (size_t)(rowbase + rg * 16 + ro) * (T_STEPS * 512) + gcol;
  }
  const float* hrow = hs + (rg * 16 + l16) * HS_STRIDE + khalf;

  // preload t=0 gx tile (bias already folded in by kernel 1)
  v8f gbuf[8];
#pragma unroll
  for (int j = 0; j < 8; ++j) {
    const float* p = gp[j];
#pragma unroll
    for (int r = 0; r < 8; ++r) gbuf[j][r] = p[r * (T_STEPS * 512)];
  }

  float cst[2][8];                          // cell state, in registers
#pragma unroll
  for (int s = 0; s < 2; ++s)
#pragma unroll
    for (int r = 0; r < 8; ++r) cst[s][r] = 0.f;

  for (int t = 0; t < T_STEPS; ++t) {
    // peel K-step 0: WMMA consumes gbuf directly as the C operand
    v8f acc[8];
    {
      v2f a0 = *(const v2f*)(hrow);
      v2f b0[8];
#pragma unroll
      for (int j = 0; j < 8; ++j) b0[j] = *(const v2f*)(bb[j]);
#pragma unroll
      for (int j = 0; j < 8; ++j) acc[j] = WMMA4(a0, b0[j], gbuf[j]);
    }
    // issue t+1 gx loads now; they overlap the remaining GEMM + cell update
    {
      int tn = (t + 1 < T_STEPS) ? (t + 1) : t;   // harmless reload on last t
#pragma unroll
      for (int j = 0; j < 8; ++j) {
        const float* p = gp[j] + (size_t)tn * 512;
#pragma unroll
        for (int r = 0; r < 8; ++r) gbuf[j][r] = p[r * (T_STEPS * 512)];
      }
    }
    // remaining 31 K-steps (4..124)
    wmma_kloop<HID_C, 4>(hrow, bb, acc);
    __syncthreads();   // all hs reads complete before overwrite

    // cell update fully in registers (i,f,g,o tiles co-resident in this wave)
#pragma unroll
    for (int s = 0; s < 2; ++s) {
      int hcol = cw * 32 + s * 16 + l16;
#pragma unroll
      for (int r = 0; r < 8; ++r) {
        float i_ = 1.f / (1.f + __expf(-acc[0 + s][r]));
        float f_ = 1.f / (1.f + __expf(-acc[2 + s][r]));
        float g_ = tanhf(acc[4 + s][r]);
        float o_ = 1.f / (1.f + __expf(-acc[6 + s][r]));
        float c_ = f_ * cst[s][r] + i_ * g_;
        cst[s][r] = c_;
        hs[(rg * 16 + r + ro) * HS_STRIDE + hcol] = o_ * tanhf(c_);
      }
    }
    __syncthreads();
  }

  // write h_last (coalesced)
  for (int i = tid; i < R2_ROWS * HID_C; i += 256) {
    int r = i >> 7, c = i & 127;
    h_last[(size_t)(rowbase + r) * HID_C + c] = hs[r * HS_STRIDE + c];
  }
}

// =============== Kernel 3: fc2 + concat + outer product ===============
#define B_ROWS 64
#define YS_STRIDE 132

__launch_bounds__(128)
__global__ void head_kernel(const float* __restrict__ cluster_num,
                            const float* __restrict__ pref,
                            const float* __restrict__ fc_w,
                            const float* __restrict__ fc_b,
                            const float* __restrict__ h_last,
                            float* __restrict__ out) {
  __shared__ float ys[B_ROWS * YS_STRIDE];
  const int tid   = threadIdx.x;
  const int lane  = tid & 31;
  const int wave  = tid >> 5;
  const int l16   = lane & 15;
  const int khalf = (lane >> 4) << 1;
  const int ro    = (lane >> 4) << 3;
  const int blockrow = blockIdx.x * B_ROWS;
  const int waverow  = wave * 16;

  const float* fw[8];
  v8f acc[8];
#pragma unroll
  for (int j = 0; j < 8; ++j) {
    int n = 16 * j + l16;
    fw[j] = fc_w + (size_t)n * EDGE_C + khalf;
    float bv = fc_b[n];
#pragma unroll
    for (int r = 0; r < 8; ++r) acc[j][r] = bv;
  }
  const float* arow =
      cluster_num + (size_t)(blockrow + waverow + l16) * EDGE_C + khalf;

  wmma_kloop<EDGE_C, 0>(arow, fw, acc);   // 8 K-steps

#pragma unroll
  for (int j = 0; j < 8; ++j) {
#pragma unroll
    for (int r = 0; r < 8; ++r) {
      float v = acc[j][r];
      ys[(waverow + r + ro) * YS_STRIDE + 16 * j + l16] = v > 0.f ? v : 0.f;
    }
  }
  __syncthreads();

  for (int lb = 0; lb < B_ROWS; ++lb) {
    int b = blockrow + lb;
    float ya = h_last[(size_t)b * HID_C + tid];
    float yb = ys[lb * YS_STRIDE + tid];
    const float* pb = pref + (size_t)b * (TPC_C * 3);
#pragma unroll
    for (int l = 0; l < TPC_C; ++l) {
      float p0 = pb[l * 3 + 0], p1 = pb[l * 3 + 1], p2 = pb[l * 3 + 2];
      float* ob = out + ((size_t)b * TPC_C + l) * 768;
      float* oa = ob + tid * 3;
      oa[0] = ya * p0; oa[1] = ya * p1; oa[2] = ya * p2;
      float* oc = ob + (128 + tid) * 3;
      oc[0] = yb * p0; oc[1] = yb * p1; oc[2] = yb * p2;
    }
  }
}

extern "C" void kernel_launch(void* const* d_in, const int* in_sizes, int n_in,
                              void* d_out, int out_size, void* d_ws, size_t ws_size,
                              hipStream_t stream) {
  const float* history = (const float*)d_in[0];
  const float* cluster = (const float*)d_in[1];
  const float* pref    = (const float*)d_in[2];
  const float* W_ih    = (const float*)d_in[3];
  const float* W_hh    = (const float*)d_in[4];
  const float* b_ih    = (const float*)d_in[5];
  const float* b_hh    = (const float*)d_in[6];
  const float* fc_w    = (const float*)d_in[7];
  const float* fc_b    = (const float*)d_in[8];
  float* out = (float*)d_out;

  // workspace: gx [163840 x 512] f32 (336 MB) followed by h_last [8192 x 128]
  float* gx     = (float*)d_ws;
  float* h_last = gx + (size_t)MROWS * 512;

  // allow >64KB dynamic LDS for the recurrence kernel (host-side, not captured)
  (void)hipFuncSetAttribute((const void*)lstm_recur_kernel,
                            hipFuncAttributeMaxDynamicSharedMemorySize,
                            SMEM2_BYTES);

  gx_gemm_kernel<<<MROWS / G1_ROWS, 256, 0, stream>>>(history, W_ih, b_ih, b_hh, gx);
  lstm_recur_kernel<<<B_TOTAL / R2_ROWS, 256, SMEM2_BYTES, stream>>>(gx, W_hh, h_last);
  head_kernel<<<B_TOTAL / B_ROWS, 128, 0, stream>>>(cluster, pref, fc_w, fc_b,
                                                    h_last, out);
}